// ChronoRotationTransflormation_5179730559619
// MI455X (gfx1250) — compile-verified
//
#include <hip/hip_runtime.h>

// ChronoRotationTransformation: per-row complex-rotation cosine similarity.
//   rot_r = hr*rr - hi*ri ; rot_i = -(hi*rr + hr*ri)
//   ab = sum(rot_r*tr + rot_i*ti); aa = sum(rot_r^2 + rot_i^2); bb = sum(tr^2 + ti^2)
//   out = ab / sqrt(aa*bb)
//
// Roofline: 6 arrays x 8192 x 2048 fp32 = 384 MB read once, 32 KB written ->
// ~16.5 us floor at 23.3 TB/s; arithmetic intensity ~0.9 FLOP/B, far under the
// VALU ceiling. Pure HBM streaming kernel:
//  - native-vector float4 (b128) fully-coalesced loads, non-temporal (TH=NT)
//    since every byte is used exactly once and the 384 MB stream exceeds the
//    192 MB L2
//  - per-lane FMA accumulation, 8 wave32s per block, one row per wave
//  - finish with V_WMMA_F32_16X16X4_F32 (B = ones) reducing all 8 block rows'
//    partials in one matrix op per quantity (3 WMMAs/block)

typedef __attribute__((ext_vector_type(2))) float v2f;
typedef __attribute__((ext_vector_type(4))) float v4f;   // native clang vector
typedef __attribute__((ext_vector_type(8))) float v8f;

#define ROWS_PER_BLOCK 8

__global__ __launch_bounds__(256)
void chrono_rot_cosine_kernel(const float* __restrict__ g_hr,
                              const float* __restrict__ g_hi,
                              const float* __restrict__ g_rr,
                              const float* __restrict__ g_ri,
                              const float* __restrict__ g_tr,
                              const float* __restrict__ g_ti,
                              float* __restrict__ g_out,
                              int D, int rows) {
    const int tid  = threadIdx.x;
    const int wave = tid >> 5;          // 8 waves per block, one row per wave
    const int lane = tid & 31;
    const int row  = blockIdx.x * ROWS_PER_BLOCK + wave;

    // Per-lane partial accumulators.
    float ab = 0.0f, aa = 0.0f, bb = 0.0f;

    if (row < rows) {
        const v4f* hr4 = reinterpret_cast<const v4f*>(g_hr);
        const v4f* hi4 = reinterpret_cast<const v4f*>(g_hi);
        const v4f* rr4 = reinterpret_cast<const v4f*>(g_rr);
        const v4f* ri4 = reinterpret_cast<const v4f*>(g_ri);
        const v4f* tr4 = reinterpret_cast<const v4f*>(g_tr);
        const v4f* ti4 = reinterpret_cast<const v4f*>(g_ti);

        const int    d4    = D >> 2;                 // float4s per row
        const int    iters = D >> 7;                 // 32 lanes * 4 floats / iter
        const size_t rowb4 = (size_t)row * (size_t)d4;

        #pragma unroll 2
        for (int i = 0; i < iters; ++i) {
            const size_t idx4 = rowb4 + (size_t)i * 32 + lane;

            // Streamed exactly once -> non-temporal loads (TH=NT), don't
            // rinse the 192MB L2 with a 384MB one-shot stream.
            const v4f hr = __builtin_nontemporal_load(hr4 + idx4);
            const v4f hi = __builtin_nontemporal_load(hi4 + idx4);
            const v4f rr = __builtin_nontemporal_load(rr4 + idx4);
            const v4f ri = __builtin_nontemporal_load(ri4 + idx4);
            const v4f tr = __builtin_nontemporal_load(tr4 + idx4);
            const v4f ti = __builtin_nontemporal_load(ti4 + idx4);

#define ACC(c)                                                            \
            {                                                             \
                float rot_r = fmaf(hr.c, rr.c, -(hi.c * ri.c));           \
                float rot_i = -fmaf(hi.c, rr.c, hr.c * ri.c);             \
                ab = fmaf(rot_r, tr.c, fmaf(rot_i, ti.c, ab));            \
                aa = fmaf(rot_r, rot_r, fmaf(rot_i, rot_i, aa));          \
                bb = fmaf(tr.c, tr.c, fmaf(ti.c, ti.c, bb));              \
            }
            ACC(x) ACC(y) ACC(z) ACC(w)
#undef ACC
        }
    }

    // ---- Wave-level pre-reduce: 32 lane partials -> 4 group partials ----
    // After these steps lanes 0..3 hold sums over {l, l+4, ..., l+28}.
    ab += __shfl_down(ab, 16, 32);
    aa += __shfl_down(aa, 16, 32);
    bb += __shfl_down(bb, 16, 32);
    ab += __shfl_down(ab, 8, 32);
    aa += __shfl_down(aa, 8, 32);
    bb += __shfl_down(bb, 8, 32);
    ab += __shfl_down(ab, 4, 32);
    aa += __shfl_down(aa, 4, 32);
    bb += __shfl_down(bb, 4, 32);

    __shared__ float s_ab[ROWS_PER_BLOCK][4];
    __shared__ float s_aa[ROWS_PER_BLOCK][4];
    __shared__ float s_bb[ROWS_PER_BLOCK][4];
    if (lane < 4) {
        s_ab[wave][lane] = ab;
        s_aa[wave][lane] = aa;
        s_bb[wave][lane] = bb;
    }
    __syncthreads();

    // ---- Final reduction on the matrix core: D = A(16x4) x ones(4x16) ----
    // A-matrix f32 16x4 VGPR layout (wave32): lane L in [0,16): {A[L][0],A[L][1]},
    // lane L+16: {A[L][2],A[L][3]}. With B = ones, D[m][n] = rowsum(m) and the
    // D layout puts row j's sum in v8f component j for lanes 0..15.
    if (tid < 32) {                       // wave 0 only; full EXEC within wave
        const int m  = tid & 15;
        const int kh = (tid >> 4) & 1;    // which half of K this lane supplies

        v2f a_ab, a_aa, a_bb;
        if (m < ROWS_PER_BLOCK) {
            a_ab = (v2f){s_ab[m][kh * 2 + 0], s_ab[m][kh * 2 + 1]};
            a_aa = (v2f){s_aa[m][kh * 2 + 0], s_aa[m][kh * 2 + 1]};
            a_bb = (v2f){s_bb[m][kh * 2 + 0], s_bb[m][kh * 2 + 1]};
        } else {
            a_ab = (v2f){0.0f, 0.0f};
            a_aa = (v2f){0.0f, 0.0f};
            a_bb = (v2f){0.0f, 0.0f};
        }
        const v2f b_ones = (v2f){1.0f, 1.0f};
        v8f c0 = {};

        v8f d_ab = __builtin_amdgcn_wmma_f32_16x16x4_f32(
            false, a_ab, false, b_ones, (short)0, c0, false, false);
        v8f d_aa = __builtin_amdgcn_wmma_f32_16x16x4_f32(
            false, a_aa, false, b_ones, (short)0, c0, false, false);
        v8f d_bb = __builtin_amdgcn_wmma_f32_16x16x4_f32(
            false, a_bb, false, b_ones, (short)0, c0, false, false);

        if (tid == 0) {
            const int base = blockIdx.x * ROWS_PER_BLOCK;
            #pragma unroll
            for (int j = 0; j < ROWS_PER_BLOCK; ++j) {
                if (base + j < rows) {
                    g_out[base + j] = d_ab[j] * __frsqrt_rn(d_aa[j] * d_bb[j]);
                }
            }
        }
    }
}

extern "C" void kernel_launch(void* const* d_in, const int* in_sizes, int n_in,
                              void* d_out, int out_size, void* d_ws, size_t ws_size,
                              hipStream_t stream) {
    (void)n_in; (void)d_ws; (void)ws_size;
    const float* hr = (const float*)d_in[0];  // head_real
    const float* hi = (const float*)d_in[1];  // head_imag
    const float* rr = (const float*)d_in[2];  // rel_real
    const float* ri = (const float*)d_in[3];  // rel_imag
    const float* tr = (const float*)d_in[4];  // tail_real
    const float* ti = (const float*)d_in[5];  // tail_imag
    float* out = (float*)d_out;

    const int rows = out_size;               // B = 8192
    const int D    = in_sizes[0] / rows;     // D = 2048

    const int blocks = (rows + ROWS_PER_BLOCK - 1) / ROWS_PER_BLOCK;
    chrono_rot_cosine_kernel<<<blocks, 256, 0, stream>>>(hr, hi, rr, ri, tr, ti,
                                                         out, D, rows);
}